// RCOptimModel_31215822307906
// MI455X (gfx1250) — compile-verified
//
#include <hip/hip_runtime.h>

// ---------------------------------------------------------------------------
// RK4(3/8) of dx/dt = A x + B u(t), N=512, 19999 fixed steps of dt=30.
// Restructured as x_{n+1} = P x_n + g_n  (P = deg-4 Taylor of exp(dt A)),
// then block-parallelized with S=16:
//   g_n      : rank-5 combination of interpolated Tout at 4 stage times
//   D-sweep  : 16 WMMA GEMMs -> coarse forcing  C_k = sum_j P^{15-j} g_{16k+j}
//   scan     : Y_{k+1} = P^16 Y_k + C_k   (1249 iters, 32 WGs, LDS-resident P^16)
//   X-sweep  : 16 WMMA GEMMs X_s = P X_{s-1} + G_{s-1}, streaming all outputs
// All f32 via V_WMMA_F32_16X16X4_F32 (native fp32 matrix core on CDNA5).
// GEMM: 2 column-tiles per wave (A-fragment reuse), branch-free inner loop
// (OOB columns clamped at load; their results feed only unstored columns).
// Workspace requirement: ~53 MB.
// ---------------------------------------------------------------------------

#define NN 512
#define TSTEPS 20000
#define NSTEP 19999          // number of integration steps
#define SBLK 16              // steps per coarse block
#define KBLK 1250            // coarse blocks (covers a guarded phantom step)
#define NDATA 2000
#define DTSTEP 30.0f
#define NWG_SCAN 32
#define SCAN_ITERS 1249      // k = 0..1248 -> Y_1..Y_1249

typedef float v2f __attribute__((ext_vector_type(2)));
typedef float v8f __attribute__((ext_vector_type(8)));

struct GemmArgs {
  const float* A; int lda;                   // A(i,j) = A[i*lda + j], M=K=512
  const float* B; int bsj, bsk;              // B(j,k) = B[j*bsj + k*bsk]
  const float* Add; int addoff, addsi, addsk;// optional addend (affine index)
  float alpha; int addIdent;
  float* D0; int d0si, d0sk;                 // primary store
  float* D1; int d1off, d1si, d1sk;          // optional second store (to d_out)
  int gmul, gadd, gmax;                      // guard: store1 iff col*gmul+gadd < gmax
  int Ndim;
};

// ---- fp32 WMMA GEMM: D = alpha*A@B (+I) (+Add) -----------------------------
// One 16-row tile x two 16-col tiles per wave; A fragment shared by both WMMAs.
__global__ void gemm_wmma_f32(GemmArgs g) {
  const int lane    = threadIdx.x & 31;
  const int wave    = (blockIdx.x * blockDim.x + threadIdx.x) >> 5;
  const int tilesM  = NN / 16;
  const int tilesN2 = (g.Ndim + 31) >> 5;     // pairs of 16-wide column tiles
  if (wave >= tilesM * tilesN2) return;       // whole-wave exit: EXEC all-1s
  const int tm   = wave % tilesM;
  const int tn2  = wave / tilesM;
  const int half = lane >> 4;                 // 0: lanes 0-15, 1: lanes 16-31
  const int l15  = lane & 15;
  const int row  = tm * 16 + l15;             // A-fragment row (M)
  const int c0   = tn2 * 32 + l15;            // first column tile
  const int c1   = c0 + 16;                   // second column tile
  // Clamp load columns: a clamped lane's B data feeds only its own (unstored)
  // output column, so valid columns are never contaminated.
  const int c0L  = c0 < g.Ndim ? c0 : g.Ndim - 1;
  const int c1L  = c1 < g.Ndim ? c1 : g.Ndim - 1;
  const float* Arow = g.A + (long)row * g.lda;
  const float* B0   = g.B + (long)c0L * g.bsk;
  const float* B1   = g.B + (long)c1L * g.bsk;
  const long  bsj   = g.bsj;

  v8f acc0 = {}, acc1 = {};
  for (int j = 0; j < NN; j += 4) {
    const int jk = j + 2 * half;
    // A 16x4 f32 layout: lanes 0-15 hold K={j,j+1}; lanes 16-31 K={j+2,j+3}
    v2f a = *(const v2f*)(Arow + jk);
    // B 4x16 f32 layout: VGPR0 lanes0-15 K=j, lanes16-31 K=j+2; VGPR1 K=j+1/j+3
    v2f b0, b1;
    b0.x = B0[(long)jk * bsj];  b0.y = B0[(long)(jk + 1) * bsj];
    b1.x = B1[(long)jk * bsj];  b1.y = B1[(long)(jk + 1) * bsj];
    acc0 = __builtin_amdgcn_wmma_f32_16x16x4_f32(false, a, false, b0,
                                                 (short)0, acc0, false, false);
    acc1 = __builtin_amdgcn_wmma_f32_16x16x4_f32(false, a, false, b1,
                                                 (short)0, acc1, false, false);
  }
  #pragma unroll
  for (int r = 0; r < 8; ++r) {
    // C/D layout: VGPR r -> row tm*16 + r + 8*half, col = lane&15
    const int orow = tm * 16 + r + 8 * half;
    if (c0 < g.Ndim) {
      float v = g.alpha * acc0[r];
      if (g.addIdent && orow == c0) v += 1.0f;
      if (g.Add) v += g.Add[(long)g.addoff + (long)orow * g.addsi + (long)c0 * g.addsk];
      g.D0[(long)orow * g.d0si + (long)c0 * g.d0sk] = v;
      if (g.D1 && (c0 * g.gmul + g.gadd) < g.gmax)
        g.D1[(long)g.d1off + (long)orow * g.d1si + (long)c0 * g.d1sk] = v;
    }
    if (c1 < g.Ndim) {
      float v = g.alpha * acc1[r];
      if (g.addIdent && orow == c1) v += 1.0f;
      if (g.Add) v += g.Add[(long)g.addoff + (long)orow * g.addsi + (long)c1 * g.addsk];
      g.D0[(long)orow * g.d0si + (long)c1 * g.d0sk] = v;
      if (g.D1 && (c1 * g.gmul + g.gadd) < g.gmax)
        g.D1[(long)g.d1off + (long)orow * g.d1si + (long)c1 * g.d1sk] = v;
    }
  }
}

// ---- per-launch reinit: zero D0 + sync flag, seed out[0]=Y_0=iv ------------
__global__ void k_init(float* D0, float* out, float* Y, const float* iv,
                       unsigned int* flag) {
  const int idx = blockIdx.x * blockDim.x + threadIdx.x;
  if (idx < NN * KBLK) D0[idx] = 0.0f;
  if (idx < NN) { out[idx] = iv[idx]; Y[idx] = iv[idx]; }
  if (idx == 0) *flag = 0u;
}

// ---- forcing basis vectors v1..v4, c (six A-matvecs, one workgroup) --------
__global__ void k_vecs(const float* A, const float* Bm, const float* Q,
                       float* V1, float* V2, float* V3, float* V4, float* CV) {
  __shared__ float cur[NN];
  const int i = threadIdx.x;
  const float h = DTSTEP, h2 = h * h, h3 = h2 * h, c8 = h / 8.0f;
  const float b0 = Bm[i * 9];
  float bq = 0.0f;
  for (int j = 0; j < 8; ++j) bq += Bm[i * 9 + 1 + j] * Q[j];
  const float* Ar = A + (long)i * NN;
  float r1, r2, r3, s1, s2, s3;

  cur[i] = b0; __syncthreads();
  r1 = 0.f; for (int j = 0; j < NN; ++j) r1 += Ar[j] * cur[j];
  __syncthreads(); cur[i] = r1; __syncthreads();
  r2 = 0.f; for (int j = 0; j < NN; ++j) r2 += Ar[j] * cur[j];
  __syncthreads(); cur[i] = r2; __syncthreads();
  r3 = 0.f; for (int j = 0; j < NN; ++j) r3 += Ar[j] * cur[j];
  __syncthreads(); cur[i] = bq; __syncthreads();
  s1 = 0.f; for (int j = 0; j < NN; ++j) s1 += Ar[j] * cur[j];
  __syncthreads(); cur[i] = s1; __syncthreads();
  s2 = 0.f; for (int j = 0; j < NN; ++j) s2 += Ar[j] * cur[j];
  __syncthreads(); cur[i] = s2; __syncthreads();
  s3 = 0.f; for (int j = 0; j < NN; ++j) s3 += Ar[j] * cur[j];

  V1[i] = c8 * (b0 + h * r1 + (h2 / 3.f) * r2 + (h3 / 3.f) * r3);
  V2[i] = c8 * (3.f * b0 + 2.f * h * r1 + h2 * r2);
  V3[i] = c8 * (3.f * b0 + h * r1);
  V4[i] = c8 * b0;
  CV[i] = c8 * (8.f * bq + 4.f * h * s1 + (4.f / 3.f) * h2 * s2 + (h3 / 3.f) * s3);
}

// ---- dst = I + alpha*src (Horner seed) -------------------------------------
__global__ void k_axpI(float* dst, const float* src, float alpha) {
  const int idx = blockIdx.x * blockDim.x + threadIdx.x;
  if (idx >= NN * NN) return;
  const int r = idx / NN, c = idx % NN;
  dst[idx] = alpha * src[idx] + (r == c ? 1.0f : 0.0f);
}

// ---- G[n] = sum_s Tout(t_n + 10 s) * v_{s+1} + c ; G[19999] = 0 ------------
__global__ void k_G(float* G, const float* Tt, const float* Tv,
                    const float* V1, const float* V2, const float* V3,
                    const float* V4, const float* CV) {
  const int n = blockIdx.x, i = threadIdx.x;
  float* grow = G + (long)n * NN;
  if (n >= NSTEP) { grow[i] = 0.0f; return; }
  const float t0 = (float)n * DTSTEP;
  float tv[4];
  #pragma unroll
  for (int s = 0; s < 4; ++s) {
    const float tt = t0 + 10.0f * (float)s;      // stage times t, t+h/3, t+2h/3, t+h
    float pos = tt * ((float)(NDATA - 1) / 600000.0f); // Tout_t is uniform linspace
    int j = (int)pos;
    j = j < 0 ? 0 : (j > NDATA - 2 ? NDATA - 2 : j);
    const float x0 = Tt[j], x1 = Tt[j + 1];
    float f = (tt - x0) / (x1 - x0);
    f = fminf(fmaxf(f, 0.0f), 1.0f);
    tv[s] = Tv[j] + f * (Tv[j + 1] - Tv[j]);
  }
  grow[i] = tv[0] * V1[i] + tv[1] * V2[i] + tv[2] * V3[i] + tv[3] * V4[i] + CV[i];
}

// ---- coarse scan: Y_{k+1} = P16 Y_k + C_k. 32 WGs, LDS slab of P16 ---------
__global__ void k_scan(const float* P16, float* Y, const float* Cfor,
                       unsigned int* flag) {
  __shared__ float slab[16 * NN];   // 16 rows of P^16 (32 KB)
  __shared__ float ycur[NN];
  const int tid = threadIdx.x;      // 256 threads
  const int w   = blockIdx.x;       // 0..31 -> rows [16w, 16w+16)
  for (int idx = tid; idx < 16 * NN; idx += 256)
    slab[idx] = P16[(long)(w * 16 + (idx >> 9)) * NN + (idx & (NN - 1))];
  const int lr  = tid >> 4;         // local row 0..15
  const int sub = tid & 15;         // 16 threads per row
  const int r   = w * 16 + lr;
  for (int k = 0; k < SCAN_ITERS; ++k) {
    __syncthreads();
    for (int idx = tid; idx < NN; idx += 256) ycur[idx] = Y[(long)k * NN + idx];
    __syncthreads();
    float s = 0.0f;
    const float* srow = slab + lr * NN + sub * 32;
    const float* yv   = ycur + sub * 32;
    #pragma unroll 8
    for (int j = 0; j < 32; ++j) s += srow[j] * yv[j];
    #pragma unroll
    for (int m = 1; m < 16; m <<= 1) s += __shfl_xor(s, m, 32);  // wave32
    if (sub == 0) Y[(long)(k + 1) * NN + r] = s + Cfor[(long)r * KBLK + k];
    __threadfence();
    __syncthreads();
    if (tid == 0) {
      __hip_atomic_fetch_add(flag, 1u, __ATOMIC_ACQ_REL, __HIP_MEMORY_SCOPE_AGENT);
      const unsigned target = (unsigned)NWG_SCAN * (unsigned)(k + 1);
      while (__hip_atomic_load(flag, __ATOMIC_ACQUIRE, __HIP_MEMORY_SCOPE_AGENT) < target)
        __builtin_amdgcn_s_sleep(1);
    }
    __syncthreads();
  }
}

extern "C" void kernel_launch(void* const* d_in, const int* in_sizes, int n_in,
                              void* d_out, int out_size, void* d_ws, size_t ws_size,
                              hipStream_t stream) {
  const float* A  = (const float*)d_in[0];   // (512,512)
  const float* Bm = (const float*)d_in[1];   // (512,9)
  const float* Q  = (const float*)d_in[2];   // (8,)
  const float* Tt = (const float*)d_in[3];   // (2000,)
  const float* Tv = (const float*)d_in[4];   // (2000,)
  const float* iv = (const float*)d_in[5];   // (512,)
  float* out = (float*)d_out;                // (20000,512)
  float* w   = (float*)d_ws;

  float* P   = w;                 // 512*512
  float* P16 = P   + NN * NN;
  float* T0  = P16 + NN * NN;
  float* T1  = T0  + NN * NN;
  float* V1  = T1  + NN * NN;
  float* V2  = V1 + NN; float* V3 = V2 + NN; float* V4 = V3 + NN; float* CV = V4 + NN;
  float* G   = CV + NN;                       // 20000*512
  float* D0  = G  + (long)TSTEPS * NN;        // 512*1250
  float* D1  = D0 + NN * KBLK;                // 512*1250
  float* Y   = D1 + NN * KBLK;                // 1250*512
  unsigned int* flag = (unsigned int*)(Y + NN * KBLK);

  const float h = DTSTEP;

  k_init<<<(NN * KBLK + 255) / 256, 256, 0, stream>>>(D0, out, Y, iv, flag);
  k_vecs<<<1, NN, 0, stream>>>(A, Bm, Q, V1, V2, V3, V4, CV);
  k_axpI<<<(NN * NN + 255) / 256, 256, 0, stream>>>(T0, A, h / 4.0f);

  auto gemm = [&](const float* Am, const float* Bp, int bsj, int bsk,
                  const float* Add, int addoff, int addsi, int addsk,
                  float alpha, int ident,
                  float* Dst0, int d0si, int d0sk,
                  float* Dst1, int d1off, int d1si, int d1sk,
                  int gmul, int gadd, int gmax, int Ndim) {
    GemmArgs ga;
    ga.A = Am; ga.lda = NN;
    ga.B = Bp; ga.bsj = bsj; ga.bsk = bsk;
    ga.Add = Add; ga.addoff = addoff; ga.addsi = addsi; ga.addsk = addsk;
    ga.alpha = alpha; ga.addIdent = ident;
    ga.D0 = Dst0; ga.d0si = d0si; ga.d0sk = d0sk;
    ga.D1 = Dst1; ga.d1off = d1off; ga.d1si = d1si; ga.d1sk = d1sk;
    ga.gmul = gmul; ga.gadd = gadd; ga.gmax = gmax;
    ga.Ndim = Ndim;
    const int tiles  = (NN / 16) * ((Ndim + 31) / 32);  // waves (2 col-tiles each)
    const int blocks = (tiles * 32 + 255) / 256;
    gemm_wmma_f32<<<blocks, 256, 0, stream>>>(ga);
  };

  // P = I + hA(I + (h/2)A(I + (h/3)A(I + (h/4)A)))  (Horner, T0 pre-seeded)
  gemm(A, T0, NN, 1, nullptr, 0, 0, 0, h / 3.f, 1, T1, NN, 1, nullptr, 0, 0, 0, 0, 0, 0, NN);
  gemm(A, T1, NN, 1, nullptr, 0, 0, 0, h / 2.f, 1, T0, NN, 1, nullptr, 0, 0, 0, 0, 0, 0, NN);
  gemm(A, T0, NN, 1, nullptr, 0, 0, 0, h,       1, P,  NN, 1, nullptr, 0, 0, 0, 0, 0, 0, NN);
  // P^16 by repeated squaring
  gemm(P,  P,  NN, 1, nullptr, 0, 0, 0, 1.f, 0, T0,  NN, 1, nullptr, 0, 0, 0, 0, 0, 0, NN);
  gemm(T0, T0, NN, 1, nullptr, 0, 0, 0, 1.f, 0, T1,  NN, 1, nullptr, 0, 0, 0, 0, 0, 0, NN);
  gemm(T1, T1, NN, 1, nullptr, 0, 0, 0, 1.f, 0, T0,  NN, 1, nullptr, 0, 0, 0, 0, 0, 0, NN);
  gemm(T0, T0, NN, 1, nullptr, 0, 0, 0, 1.f, 0, P16, NN, 1, nullptr, 0, 0, 0, 0, 0, 0, NN);

  k_G<<<TSTEPS, NN, 0, stream>>>(G, Tt, Tv, V1, V2, V3, V4, CV);

  // D-sweep: D^(s+1) = P D^(s) + G_s  (D^(0)=0) -> coarse forcing in D0
  {
    float* cur = D0; float* nxt = D1;
    for (int s = 0; s < SBLK; ++s) {
      gemm(P, cur, KBLK, 1, G, s * NN, 1, SBLK * NN, 1.f, 0,
           nxt, KBLK, 1, nullptr, 0, 0, 0, 0, 0, 0, KBLK);
      float* t = cur; cur = nxt; nxt = t;
    }
    // 16 swaps from D0 -> result back in D0 == coarse forcing C
  }

  k_scan<<<NWG_SCAN, 256, 0, stream>>>(P16, Y, D0, flag);

  // X-sweep: X^(0)=Y; X^(s) = P X^(s-1) + G_{s-1}; col k of X^(s) is out row 16k+s
  {
    const float* Bcur = Y; int bsj = 1, bsk = NN;   // Y stored k-major
    float* dst = D1;                                 // D0 (=C) free after scan
    for (int s = 1; s <= SBLK; ++s) {
      gemm(P, Bcur, bsj, bsk, G, (s - 1) * NN, 1, SBLK * NN, 1.f, 0,
           dst, KBLK, 1,
           out, s * NN, 1, SBLK * NN, SBLK, s, TSTEPS, KBLK);
      Bcur = dst; bsj = KBLK; bsk = 1;
      dst = (dst == D1) ? D0 : D1;
    }
  }
}